// LocallyConnected2d_83683142795304
// MI455X (gfx1250) — compile-verified
//
#include <hip/hip_runtime.h>

// LocallyConnected2d on gfx1250: per-position GEMM D(8x32) = A(8x288) x B(288x32)
// using V_WMMA_F32_16X16X4_F32 (f32 matches reference precision).

typedef __attribute__((ext_vector_type(2))) float v2f;
typedef __attribute__((ext_vector_type(8))) float v8f;

#define KH 3
#define KW 3
#define DH 2
#define DW 2
#define BATCH 8
#define CIN 32
#define COUT 32
#define HIN 64
#define WIN 64
#define HO 31
#define WO 31
#define KKTOT 288          // CIN * KH * KW
#define WPITCH 33          // sW row pitch (pad vs 64-bank LDS)
#define APITCH 17          // sA row pitch

__global__ __launch_bounds__(64)
void lc2d_wmma_f32(const float* __restrict__ x,
                   const float* __restrict__ wgt,
                   float* __restrict__ out)
{
    // sW[kk*WPITCH + o] = weight[o, i, p, q, k], kk = i*9 + k   (B matrix, K x N)
    // sA[kk*APITCH + m] = x[m, i, 2p+kh, 2q+kw] (m<8), 0 (m>=8) (A matrix, M x K)
    __shared__ float sW[KKTOT * WPITCH];   // 38,016 B
    __shared__ float sA[KKTOT * APITCH];   // 19,584 B

    const int tid = threadIdx.x;
    const int p = blockIdx.x / WO;
    const int q = blockIdx.x % WO;

    // ---------- stage weight slice (k-contiguous global reads) ----------
    const long wPQ = (long)p * (WO * 9) + (long)q * 9;   // position offset inside (o,i) plane
    for (int e = tid; e < COUT * KKTOT; e += 64) {
        const int o  = e / KKTOT;
        const int kk = e - o * KKTOT;
        const int i  = kk / 9;
        const int k  = kk - i * 9;
        sW[kk * WPITCH + o] =
            wgt[(long)o * (CIN * HO * WO * 9) + (long)i * (HO * WO * 9) + wPQ + k];
    }

    // ---------- stage input windows (rows 0..7 real, 8..15 zero) ----------
    for (int e = tid; e < BATCH * KKTOT; e += 64) {
        const int m  = e / KKTOT;
        const int kk = e - m * KKTOT;
        const int i  = kk / 9;
        const int k  = kk - i * 9;
        const int kh = k / 3;
        const int kw = k - kh * 3;
        sA[kk * APITCH + m] =
            x[(long)m * (CIN * HIN * WIN) + (long)i * (HIN * WIN)
              + (DH * p + kh) * WIN + (DW * q + kw)];
        sA[kk * APITCH + 8 + m] = 0.0f;   // pad M to 16
    }
    __syncthreads();

    // ---------- WMMA main loop: 72 x v_wmma_f32_16x16x4_f32 per wave ----------
    const int lane  = tid & 31;
    const int half  = lane >> 4;          // selects K sub-pair within 4-wide step
    const int mn    = lane & 15;          // A: row M   /  B: col N
    const int obase = (tid >> 5) * 16;    // wave 0 -> o[0,16), wave 1 -> o[16,32)

    v8f c = {};
#pragma unroll 4
    for (int s = 0; s < KKTOT / 4; ++s) {
        const int kk0 = 4 * s + 2 * half;
        v2f a, b;
        a.x = sA[(kk0    ) * APITCH + mn];
        a.y = sA[(kk0 + 1) * APITCH + mn];
        b.x = sW[(kk0    ) * WPITCH + obase + mn];
        b.y = sW[(kk0 + 1) * WPITCH + obase + mn];
        // 8 args: (neg_a, A, neg_b, B, c_mod, C, reuse_a, reuse_b)
        c = __builtin_amdgcn_wmma_f32_16x16x4_f32(
                false, a, false, b, (short)0, c, false, false);
    }

    // ---------- store: C VGPR r / lanes 0..15 hold (b=r, o=obase+mn) ----------
    if (lane < 16) {
        const int o = obase + mn;
        float* dst = out + (long)o * (HO * WO) + (long)p * WO + q;
#pragma unroll
        for (int r = 0; r < BATCH; ++r)
            dst[(long)r * (COUT * HO * WO)] = c[r];
    }
}

extern "C" void kernel_launch(void* const* d_in, const int* in_sizes, int n_in,
                              void* d_out, int out_size, void* d_ws, size_t ws_size,
                              hipStream_t stream)
{
    const float* x   = (const float*)d_in[0];   // (8, 32, 64, 64) f32
    const float* wgt = (const float*)d_in[1];   // (1, 32, 32, 31, 31, 9) f32
    float* out = (float*)d_out;                 // (8, 32, 31, 31) f32

    dim3 grid(HO * WO);   // one workgroup (2 waves) per output position
    dim3 block(64);
    lc2d_wmma_f32<<<grid, block, 0, stream>>>(x, wgt, out);
}